// NeighborList_89172111000334
// MI455X (gfx1250) — compile-verified
//
#include <hip/hip_runtime.h>

// CDNA5 (gfx1250) neighbor list: pair distance matrix as tiled Gram matrix
// via V_WMMA_F32_16X16X4_F32.  ||xi-xj||^2 = |xi|^2 + |xj|^2 - 2*xi.xj.
// One wave32 per 16x16 tile of the (i,j) pair matrix, upper triangle only.
// Off-diagonal tiles take a branch-free emit path; only the 256 diagonal
// tiles run the per-element i<j guard.

typedef __attribute__((ext_vector_type(2))) float v2f;
typedef __attribute__((ext_vector_type(8))) float v8f;

#define CUTOFF2 25.0f  // 5.0^2

__global__ void nl_zero_count(int* __restrict__ n_pairs) { *n_pairs = 0; }

__global__ __launch_bounds__(32)
void nl_tile_kernel(const float* __restrict__ xyz, int N,
                    int*   __restrict__ out_i,
                    int*   __restrict__ out_j,
                    float* __restrict__ out_diff,
                    float* __restrict__ out_dist,
                    int*   __restrict__ n_pairs)
{
    const int bi = blockIdx.x;   // i tile block
    const int bj = blockIdx.y;   // j tile block
    if (bi > bj) return;         // uniform exit: only tiles that can hold i<j
    const int i0 = bi << 4;
    const int j0 = bj << 4;

    // Per-wave staging: 16 i-rows and 16 j-cols of coordinates + squared norms.
    __shared__ float sI[48];   // [row][xyz]
    __shared__ float sJ[48];   // [col][xyz]
    __shared__ float sNI[16];
    __shared__ float sNJ[16];

    const int l = threadIdx.x;   // 0..31 (wave32)
    const int g = l >> 4;        // half-wave group
    const int m = l & 15;

    {
        // lanes 0-15 stage i-rows, lanes 16-31 stage j-cols
        const int row = (g == 0) ? (i0 + m) : (j0 + m);
        const float x = xyz[3 * row + 0];
        const float y = xyz[3 * row + 1];
        const float z = xyz[3 * row + 2];
        float* sc = (g == 0) ? sI : sJ;
        float* sn = (g == 0) ? sNI : sNJ;
        sc[3 * m + 0] = x;
        sc[3 * m + 1] = y;
        sc[3 * m + 2] = z;
        sn[m] = x * x + y * y + z * z;
    }
    __syncthreads();

    // WMMA fragments per ISA 16x4 f32 layout (K=3 padded to 4 with zeros):
    //   lanes 0-15 hold K=0,1 in v0,v1 ; lanes 16-31 hold K=2,3.
    v2f a, b;
    a.x = (g == 0) ? sI[3 * m + 0] : sI[3 * m + 2];
    a.y = (g == 0) ? sI[3 * m + 1] : 0.0f;
    b.x = (g == 0) ? sJ[3 * m + 0] : sJ[3 * m + 2];
    b.y = (g == 0) ? sJ[3 * m + 1] : 0.0f;

    v8f c = {};
    // D = A(16x4) x B(4x16): all 256 dot products xi.xj of the tile.
    // EXEC is all-1s here (no divergence yet), as WMMA requires.
    c = __builtin_amdgcn_wmma_f32_16x16x4_f32(false, a, false, b,
                                              (short)0, c, false, false);

    // C/D layout: vgpr r -> row (g==0 ? r : r+8), col = lane&15.
    const int      n  = m;        // tile column handled by this lane
    const int      j  = j0 + n;
    const float    xj = sJ[3 * n + 0];
    const float    yj = sJ[3 * n + 1];
    const float    zj = sJ[3 * n + 2];
    const float    nj = sNJ[n];
    const unsigned uN = (unsigned)N;
    const unsigned uj = (unsigned)j;

    unsigned cnt = 0;

    if (bi < bj) {
        // ---- Fast path: strictly-above-diagonal tile, every element valid.
        // No per-element EXEC manipulation, pure streaming stores.
        #pragma unroll
        for (int r = 0; r < 8; ++r) {
            const int      mrow = r + (g << 3);
            const unsigned ui   = (unsigned)(i0 + mrow);

            const float r2  = sNI[mrow] + nj - 2.0f * c[r];
            const bool  hit = r2 < CUTOFF2;
            cnt += (unsigned)__popcll(__ballot(hit));

            // p = i*N - i*(i+1)/2 + (j-i-1); P < 2^24 so 32-bit math suffices.
            const unsigned p  = ui * uN - ((ui * (ui + 1u)) >> 1) + uj - ui - 1u;
            const unsigned p3 = 3u * p;

            const float dx = sI[3 * mrow + 0] - xj;
            const float dy = sI[3 * mrow + 1] - yj;
            const float dz = sI[3 * mrow + 2] - zj;
            const float d  = hit ? __builtin_amdgcn_sqrtf(fmaxf(r2, 0.0f)) : 0.0f;

            __builtin_nontemporal_store(hit ? (int)ui : -1, out_i + p);
            __builtin_nontemporal_store(hit ? j : -1, out_j + p);
            __builtin_nontemporal_store(hit ? dx : 0.0f, out_diff + p3 + 0);
            __builtin_nontemporal_store(hit ? dy : 0.0f, out_diff + p3 + 1);
            __builtin_nontemporal_store(hit ? dz : 0.0f, out_diff + p3 + 2);
            __builtin_nontemporal_store(d, out_dist + p);
        }
    } else {
        // ---- Diagonal tile (bi == bj): per-element strict upper-triangle guard.
        #pragma unroll
        for (int r = 0; r < 8; ++r) {
            const int      mrow = r + (g << 3);
            const unsigned ui   = (unsigned)(i0 + mrow);
            const bool     emit = (int)ui < j;

            const float r2  = sNI[mrow] + nj - 2.0f * c[r];
            const bool  hit = emit && (r2 < CUTOFF2);
            cnt += (unsigned)__popcll(__ballot(hit));

            if (emit) {
                const unsigned p  = ui * uN - ((ui * (ui + 1u)) >> 1) + uj - ui - 1u;
                const unsigned p3 = 3u * p;

                const float dx = sI[3 * mrow + 0] - xj;
                const float dy = sI[3 * mrow + 1] - yj;
                const float dz = sI[3 * mrow + 2] - zj;
                const float d  = hit ? __builtin_amdgcn_sqrtf(fmaxf(r2, 0.0f)) : 0.0f;

                __builtin_nontemporal_store(hit ? (int)ui : -1, out_i + p);
                __builtin_nontemporal_store(hit ? j : -1, out_j + p);
                __builtin_nontemporal_store(hit ? dx : 0.0f, out_diff + p3 + 0);
                __builtin_nontemporal_store(hit ? dy : 0.0f, out_diff + p3 + 1);
                __builtin_nontemporal_store(hit ? dz : 0.0f, out_diff + p3 + 2);
                __builtin_nontemporal_store(d, out_dist + p);
            }
        }
    }

    if (l == 0 && cnt) atomicAdd(n_pairs, (int)cnt);
}

extern "C" void kernel_launch(void* const* d_in, const int* in_sizes, int n_in,
                              void* d_out, int out_size, void* d_ws, size_t ws_size,
                              hipStream_t stream) {
    const float* xyz = (const float*)d_in[0];
    const int N = in_sizes[0] / 3;               // 4096
    const long long P = (long long)N * (N - 1) / 2;

    // d_out flat layout (return order): pair_i[P], pair_j[P], pair_diff[3P],
    // pair_dist[P], n_pairs[1]  (int32 values stored bit-wise in the buffer)
    int*   out_i    = (int*)d_out;
    int*   out_j    = out_i + P;
    float* out_diff = (float*)d_out + 2 * P;
    float* out_dist = (float*)d_out + 5 * P;
    int*   n_pairs  = (int*)((float*)d_out + 6 * P);

    nl_zero_count<<<1, 1, 0, stream>>>(n_pairs);

    const int T = (N + 15) / 16;                 // 256 tile blocks per dim
    dim3 grid(T, T), block(32);
    nl_tile_kernel<<<grid, block, 0, stream>>>(xyz, N, out_i, out_j,
                                               out_diff, out_dist, n_pairs);
}